// MultiHashtable4d_5952824673236
// MI455X (gfx1250) — compile-verified
//
#include <hip/hip_runtime.h>
#include <stdint.h>

#define BLK  256
#define MAXL 16

__device__ __forceinline__ int imin(int a, int b) { return a < b ? a : b; }
__device__ __forceinline__ int imax(int a, int b) { return a > b ? a : b; }

// CDNA5 async global->LDS copy (ASYNCcnt-tracked). VDST = LDS byte offset in a
// VGPR, VADDR = 64-bit global address in a VGPR pair, saddr = off.
__device__ __forceinline__ void async_ld_b32(unsigned lds_off, const void* g) {
  asm volatile("global_load_async_to_lds_b32 %0, %1, off"
               :: "v"(lds_off), "v"((unsigned long long)(uintptr_t)g)
               : "memory");
}
__device__ __forceinline__ void wait_async_0() {
  asm volatile("s_wait_asynccnt 0x0" ::: "memory");
}

// Per-level index + weight setup. Math identical to the reference:
//   fd = (pos - b0) / sz ; base = trunc(fd) ; frac = fd - base (f32)
//   corners clamped to [0, num-1]; hash = xor of int64-exact products mod T.
//
// mod T via one Barrett magic, shaped for gfx1250's native 64-bit VALU:
//   hh < 2^38 (hh_hi < 2^6), M44 = floor(2^44/T) < 2^26
//   hh*M44 = hh_lo*M44 (32x32->64 mad) + ((hh_hi*M44) << 32)   [< 2^63]
//   q = (hh*M44) >> 44  is floor(hh/T) or floor(hh/T)-1
//   true r = hh - q*T < 2T < 2^21  ->  compute in u32 wraparound:
//   r = (u32)hh - q*T (v_mul_lo); idx = min(r, r-T) unsigned.
__device__ __forceinline__ void level_setup(
    const float pos[4], const float* s_b0,
    const float* sz, const int* num, const int* str,
    bool is_hash, unsigned T, unsigned M44,
    unsigned idx[16], float wxy[2][2], float wzt[2][2])
{
  int   c0[4], c1[4];
  float w0[4], w1[4];
  #pragma unroll
  for (int d = 0; d < 4; ++d) {
    float fd   = (pos[d] - s_b0[d]) / sz[d];
    int   base = (int)fd;                 // fd >= 0 -> trunc == floor
    float frac = fd - (float)base;
    int   nm1  = num[d] - 1;
    c0[d] = imin(imax(base, 0), nm1);
    c1[d] = imin(imax(base + 1, 0), nm1);
    w1[d] = fminf(fmaxf(frac, 0.0f), 1.0f);
    w0[d] = fminf(fmaxf(1.0f - frac, 0.0f), 1.0f);
  }
  wxy[0][0] = w0[0] * w0[1]; wxy[0][1] = w0[0] * w1[1];
  wxy[1][0] = w1[0] * w0[1]; wxy[1][1] = w1[0] * w1[1];
  wzt[0][0] = w0[2] * w0[3]; wzt[0][1] = w0[2] * w1[3];
  wzt[1][0] = w1[2] * w0[3]; wzt[1][1] = w1[2] * w1[3];

  if (is_hash) {
    // per-dim int64-exact products via 32x32->64 (P0 = 1 -> hx = coord)
    unsigned long long hx[2], hy[2], hz[2], hw[2];
    #pragma unroll
    for (int j = 0; j < 2; ++j) {
      hx[j] = (unsigned long long)(unsigned)(j ? c1[0] : c0[0]);
      hy[j] = (unsigned long long)(unsigned)(j ? c1[1] : c0[1]) * 19349663u;
      hz[j] = (unsigned long long)(unsigned)(j ? c1[2] : c0[2]) * 83492791u;
      hw[j] = (unsigned long long)(unsigned)(j ? c1[3] : c0[3]) * 73856093u;
    }
    // pairwise XOR pre-combines: 2 XORs per corner afterwards
    unsigned long long hxy[2][2], hzw[2][2];
    #pragma unroll
    for (int i = 0; i < 2; ++i)
      #pragma unroll
      for (int j = 0; j < 2; ++j) {
        hxy[i][j] = hx[i] ^ hy[j];
        hzw[i][j] = hz[i] ^ hw[j];
      }
    #pragma unroll
    for (int c = 0; c < 16; ++c) {
      int bx = (c >> 3) & 1, by = (c >> 2) & 1, bz = (c >> 1) & 1, bw = c & 1;
      unsigned long long hh = hxy[bx][by] ^ hzw[bz][bw];
      unsigned hlo = (unsigned)hh;
      unsigned hhi = (unsigned)(hh >> 32);            // < 2^6
      unsigned long long prod =
          (unsigned long long)hlo * M44 +             // 32x32->64 mad
          ((unsigned long long)(hhi * M44) << 32);    // exact u32 mul
      unsigned q  = (unsigned)(prod >> 44);
      unsigned r  = hlo - q * T;                      // u32 wrap == true r
      unsigned r2 = r - T;                            // wraps huge if r < T
      idx[c] = r < r2 ? r : r2;                       // unsigned min
    }
  } else {
    int ax[2], ay[2], az[2], aw[2];
    ax[0] = c0[0] * str[0]; ax[1] = c1[0] * str[0];
    ay[0] = c0[1] * str[1]; ay[1] = c1[1] * str[1];
    az[0] = c0[2] * str[2]; az[1] = c1[2] * str[2];
    aw[0] = c0[3];          aw[1] = c1[3];
    #pragma unroll
    for (int c = 0; c < 16; ++c)
      idx[c] = (unsigned)(ax[(c >> 3) & 1] + ay[(c >> 2) & 1] +
                          az[(c >> 1) & 1] + aw[c & 1]);
  }
}

__device__ __forceinline__ float2 blend16(const float2 v[16],
                                          const float wxy[2][2],
                                          const float wzt[2][2])
{
  float ax = 0.0f, ay = 0.0f;
  #pragma unroll
  for (int c = 0; c < 16; ++c) {
    float wgt = wxy[(c >> 3) & 1][(c >> 2) & 1] * wzt[(c >> 1) & 1][c & 1];
    ax = fmaf(wgt, v[c].x, ax);
    ay = fmaf(wgt, v[c].y, ay);
  }
  float2 o; o.x = ax; o.y = ay;
  return o;
}

__global__ __launch_bounds__(BLK)
void hash4d_kernel(const float* __restrict__ xyz,
                   const float* __restrict__ tin,
                   const float* __restrict__ data,
                   const float* __restrict__ bounds,
                   const float* __restrict__ esz,
                   const int*   __restrict__ enum_,
                   const int*   __restrict__ p_start_hash,
                   const int*   __restrict__ p_start_frame,
                   const int*   __restrict__ p_total_frame,
                   float* __restrict__ out,
                   int N, int L, unsigned T, unsigned M44)
{
  __shared__ float s_sz[MAXL][4];
  __shared__ int   s_num[MAXL][4];
  __shared__ int   s_str[MAXL][3];   // strides for dims 0..2 (dim3 stride = 1)
  __shared__ float s_b0[4];
  __shared__ float s_tshift;
  __shared__ int   s_shash;
  __shared__ float s_in[BLK][4];     // staged xyzt per point (async-filled)

  const int tid = threadIdx.x;
  const int p   = blockIdx.x * BLK + tid;

  // ---- stage this block's point coords into LDS via CDNA5 async copies ----
  if (p < N) {
    unsigned off = (unsigned)(uintptr_t)&s_in[tid][0];
    const float* g = xyz + (size_t)p * 3;
    async_ld_b32(off + 0u,  g + 0);
    async_ld_b32(off + 4u,  g + 1);
    async_ld_b32(off + 8u,  g + 2);
    async_ld_b32(off + 12u, tin + p);
  }

  // ---- per-level constants -> LDS (one thread per level) ----
  if (tid < L) {
    int n[4];
    #pragma unroll
    for (int d = 0; d < 4; ++d) {
      s_sz[tid][d]  = esz[tid * 4 + d];
      n[d]          = enum_[tid * 4 + d];
      s_num[tid][d] = n[d];
    }
    s_str[tid][2] = n[3];
    s_str[tid][1] = n[2] * n[3];
    s_str[tid][0] = n[1] * n[2] * n[3];
  }
  if (tid == 0) {
    s_b0[0] = bounds[0]; s_b0[1] = bounds[1];
    s_b0[2] = bounds[2]; s_b0[3] = bounds[3];
    s_shash  = p_start_hash[0];
    s_tshift = (float)p_start_frame[0] / ((float)p_total_frame[0] - 1.0f);
  }

  wait_async_0();      // our wave's async LDS writes have landed
  __syncthreads();     // everyone's constants + staging visible

  if (p >= N) return;

  float pos[4];
  pos[0] = s_in[tid][0];
  pos[1] = s_in[tid][1];
  pos[2] = s_in[tid][2];
  pos[3] = s_in[tid][3] - s_tshift;

  const int start_hash = s_shash;
  const float2* __restrict__ tbl = (const float2*)data;            // f = 2
  float* __restrict__ orow = out + (size_t)p * (size_t)(2 * L);
  __builtin_prefetch(tbl, 0, 3);   // one global_prefetch to warm level-0 base

  // Process levels in PAIRS: compute 2x16 indices, issue 32 gathers
  // back-to-back (32 outstanding b64 loads per thread), then consume.
  // Loads return in order, so blending pair-member A only waits for the
  // first 16 loads while B's 16 remain in flight.
  for (int l = 0; l < L; l += 2) {
    const bool haveB = (l + 1) < L;

    unsigned idxA[16], idxB[16];
    float wxyA[2][2], wztA[2][2], wxyB[2][2], wztB[2][2];
    level_setup(pos, s_b0, s_sz[l], s_num[l], s_str[l],
                l >= start_hash, T, M44, idxA, wxyA, wztA);
    if (haveB)
      level_setup(pos, s_b0, s_sz[l + 1], s_num[l + 1], s_str[l + 1],
                  (l + 1) >= start_hash, T, M44, idxB, wxyB, wztB);

    const float2* ltA = tbl + (size_t)l * (size_t)T;
    const float2* ltB = tbl + (size_t)(l + 1) * (size_t)T;

    float2 vA[16], vB[16];
    #pragma unroll
    for (int c = 0; c < 16; ++c) vA[c] = ltA[idxA[c]];
    if (haveB) {
      #pragma unroll
      for (int c = 0; c < 16; ++c) vB[c] = ltB[idxB[c]];
    }

    float2 oA = blend16(vA, wxyA, wztA);
    if (haveB) {
      float2 oB = blend16(vB, wxyB, wztB);
      float4 o4; o4.x = oA.x; o4.y = oA.y; o4.z = oB.x; o4.w = oB.y;
      *(float4*)(orow + 2 * l) = o4;      // one b128 store per level pair
    } else {
      *(float2*)(orow + 2 * l) = oA;
    }
  }
}

extern "C" void kernel_launch(void* const* d_in, const int* in_sizes, int n_in,
                              void* d_out, int out_size, void* d_ws, size_t ws_size,
                              hipStream_t stream) {
  const float* xyz  = (const float*)d_in[0];
  const float* tin  = (const float*)d_in[1];
  const float* data = (const float*)d_in[2];
  const float* bnds = (const float*)d_in[3];
  // d_in[4] = offsets table: the 16 binary corners are hardcoded in-kernel
  const float* esz  = (const float*)d_in[5];
  const int*   enm  = (const int*)d_in[6];
  const int*   psh  = (const int*)d_in[7];
  const int*   psf  = (const int*)d_in[8];
  const int*   ptf  = (const int*)d_in[9];

  const int N = in_sizes[0] / 3;
  const int L = in_sizes[5] / 4;
  const int F = (N > 0 && L > 0) ? out_size / (N * L) : 2;   // = 2
  const unsigned T = (unsigned)(in_sizes[2] / (L * F));      // prime, ~2^19

  // Barrett magic: M44 = floor(2^44 / T) < 2^26 (T > 2^18)
  const unsigned M44 = (unsigned)((1ull << 44) / (unsigned long long)T);

  dim3 grid((N + BLK - 1) / BLK), block(BLK);
  hipLaunchKernelGGL(hash4d_kernel, grid, block, 0, stream,
                     xyz, tin, data, bnds, esz, enm, psh, psf, ptf,
                     (float*)d_out, N, L, T, M44);
  (void)d_ws; (void)ws_size; (void)n_in;
}